// SinkhornKnopp_89730456748862
// MI455X (gfx1250) — compile-verified
//
#include <hip/hip_runtime.h>

// Sinkhorn-Knopp, factored form: Q_final = E * r ⊗ c, E = exp(l/eps - max).
// Memory-bound (~1 GB compulsory @ 23.3 TB/s); per-slice sequencing keeps the
// 128 MB slice resident in the 192 MB L2 across the ~8 sweeps.
// WMMA f32 16x16x4 performs the GEMV rowdot[b] = sum_k E[b,k]*r[k].

#define EPS_SK  0.05f
#define NSLICE  4
#define B_DIM   8192
#define K_DIM   4096
#define ROWS    64          // rows per block in columnwise sweeps
#define KSPLIT  4           // K chunks per row-group in WMMA GEMV
#define KCH     (K_DIM / KSPLIT)

typedef float v2f __attribute__((ext_vector_type(2)));
typedef float v8f __attribute__((ext_vector_type(8)));

// workspace layout (float units)
#define WS_UMAX   0                              // [4]  monotone-uint max
#define WS_COLSUM 4                              // [4][K]
#define WS_R      (WS_COLSUM + NSLICE * K_DIM)   // [4][K]
#define WS_C      (WS_R      + NSLICE * K_DIM)   // [4][B]
#define WS_ROWDOT (WS_C      + NSLICE * B_DIM)   // [4][B]
#define WS_TOTAL  (WS_ROWDOT + NSLICE * B_DIM)

__global__ void k_init(float* ws) {
    int i = blockIdx.x * blockDim.x + threadIdx.x;
    if (i < WS_TOTAL) ws[i] = 0.0f;
}

// ---- pass 1: per-slice max of logits (monotone-uint atomicMax) ----
__global__ void k_max(const float* __restrict__ x, unsigned* __restrict__ umax, int n) {
    const float4* xv = (const float4*)(x + (size_t)n * B_DIM * K_DIM);
    const int nquads = B_DIM * K_DIM / 4;
    float m = -3.4e38f;
    for (int i = blockIdx.x * blockDim.x + threadIdx.x; i < nquads;
         i += gridDim.x * blockDim.x) {
        float4 v = xv[i];
        m = fmaxf(m, fmaxf(fmaxf(v.x, v.y), fmaxf(v.z, v.w)));
    }
    __shared__ float sm[256];
    sm[threadIdx.x] = m;
    __syncthreads();
    for (int s = 128; s > 0; s >>= 1) {
        if ((int)threadIdx.x < s) sm[threadIdx.x] = fmaxf(sm[threadIdx.x], sm[threadIdx.x + s]);
        __syncthreads();
    }
    if (threadIdx.x == 0) {
        unsigned bits = __float_as_uint(sm[0]);
        unsigned key  = (bits & 0x80000000u) ? ~bits : (bits | 0x80000000u);
        atomicMax(&umax[n], key);
    }
}

__device__ __forceinline__ float decode_max(unsigned u) {
    unsigned bits = (u & 0x80000000u) ? (u & 0x7FFFFFFFu) : ~u;
    return __uint_as_float(bits);
}

// ---- pass 2: E = exp(l/eps - M) -> d_out, colsum[k] += sum_b E ----
__global__ void k_expcol(const float* __restrict__ x, float* __restrict__ E,
                         float* __restrict__ colsum, const unsigned* __restrict__ umax,
                         int n) {
    const float M = decode_max(umax[n]) / EPS_SK;
    const size_t base = (size_t)n * B_DIM * K_DIM;
    const int k    = (blockIdx.x * blockDim.x + threadIdx.x) * 4;
    const int row0 = blockIdx.y * ROWS;
    float4 acc = {0.f, 0.f, 0.f, 0.f};
    for (int rr = 0; rr < ROWS; ++rr) {
        size_t idx = base + (size_t)(row0 + rr) * K_DIM + k;
        float4 v = *(const float4*)(x + idx);
        float4 e;
        e.x = __expf(v.x / EPS_SK - M);
        e.y = __expf(v.y / EPS_SK - M);
        e.z = __expf(v.z / EPS_SK - M);
        e.w = __expf(v.w / EPS_SK - M);
        *(float4*)(E + idx) = e;
        acc.x += e.x; acc.y += e.y; acc.z += e.z; acc.w += e.w;
    }
    float* cs = colsum + n * K_DIM + k;
    atomicAdd(cs + 0, acc.x);
    atomicAdd(cs + 1, acc.y);
    atomicAdd(cs + 2, acc.z);
    atomicAdd(cs + 3, acc.w);
}

// ---- weighted column reduction: colsum[k] += sum_b E[b,k] * c[b] ----
__global__ void k_coldot(const float* __restrict__ E, const float* __restrict__ c,
                         float* __restrict__ colsum, int n) {
    const size_t base = (size_t)n * B_DIM * K_DIM;
    const int k    = (blockIdx.x * blockDim.x + threadIdx.x) * 4;
    const int row0 = blockIdx.y * ROWS;
    float4 acc = {0.f, 0.f, 0.f, 0.f};
    for (int rr = 0; rr < ROWS; ++rr) {
        int row = row0 + rr;
        float cb = c[n * B_DIM + row];  // uniform -> scalar load
        float4 e = *(const float4*)(E + base + (size_t)row * K_DIM + k);
        acc.x += e.x * cb; acc.y += e.y * cb; acc.z += e.z * cb; acc.w += e.w * cb;
    }
    float* cs = colsum + n * K_DIM + k;
    atomicAdd(cs + 0, acc.x);
    atomicAdd(cs + 1, acc.y);
    atomicAdd(cs + 2, acc.z);
    atomicAdd(cs + 3, acc.w);
}

// ---- r[k] = 1/(K * colsum[k]); re-zero colsum for next round ----
__global__ void k_r(float* __restrict__ colsum, float* __restrict__ r, int n) {
    int k = blockIdx.x * blockDim.x + threadIdx.x;
    if (k < K_DIM) {
        float s = colsum[n * K_DIM + k];
        r[n * K_DIM + k] = 1.0f / ((float)K_DIM * s);
        colsum[n * K_DIM + k] = 0.0f;
    }
}

// ---- GEMV via WMMA: rowdot[b] += sum_k E[b,k] * r[k] ----
// A (16x4 f32): lane<16 -> row=lane, K={0,1}; lane>=16 -> row=lane-16, K={2,3}
//   => per-lane float2 load at E[row, kb + 2*(lane>>4)].
// B (4x16 f32): columns broadcast of r; row split K0/K2 across lane halves
//   => same float2 address pattern into r.
// D: all 16 columns identical; lanes with N==0 (lane 0 -> rows 0..7,
//    lane 16 -> rows 8..15) hold the row dots in their 8 accumulator VGPRs.
__global__ void k_rowdot(const float* __restrict__ E, const float* __restrict__ r,
                         float* __restrict__ rowdot, int n) {
    const int lane = threadIdx.x & 31;
    const int wave = threadIdx.x >> 5;
    const int gid  = blockIdx.x * (blockDim.x >> 5) + wave;
    const int ngroups = B_DIM / 16;          // 512 row-groups per slice
    const int grp  = gid % ngroups;
    const int kc   = gid / ngroups;          // which K chunk
    const int half = lane >> 4;
    const int mrow = lane & 15;

    const float* Es = E + (size_t)n * B_DIM * K_DIM + (size_t)(grp * 16 + mrow) * K_DIM;
    const float* rs = r + n * K_DIM;
    const int kbase0 = kc * KCH;

    v8f acc = {0.f, 0.f, 0.f, 0.f, 0.f, 0.f, 0.f, 0.f};
    for (int kb = kbase0; kb < kbase0 + KCH; kb += 4) {
        const int ko = kb + 2 * half;
        v2f a = *(const v2f*)(Es + ko);
        v2f b = *(const v2f*)(rs + ko);
        acc = __builtin_amdgcn_wmma_f32_16x16x4_f32(
            /*neg_a=*/false, a, /*neg_b=*/false, b,
            /*c_mod=*/(short)0, acc, /*reuse_a=*/false, /*reuse_b=*/false);
    }
    if (mrow == 0) {
        float* rd = rowdot + n * B_DIM + grp * 16 + half * 8;
#pragma unroll
        for (int v = 0; v < 8; ++v) atomicAdd(rd + v, acc[v]);
    }
}

// ---- c[b] = 1/(B * rowdot[b]); re-zero rowdot for next round ----
__global__ void k_c(float* __restrict__ rowdot, float* __restrict__ c, int n) {
    int b = blockIdx.x * blockDim.x + threadIdx.x;
    if (b < B_DIM) {
        float s = rowdot[n * B_DIM + b];
        c[n * B_DIM + b] = 1.0f / ((float)B_DIM * s);
        rowdot[n * B_DIM + b] = 0.0f;
    }
}

// ---- final in-place scale: out[b,k] = E[b,k] * r[k] * c[b] * B ----
__global__ void k_final(float* __restrict__ E, const float* __restrict__ r,
                        const float* __restrict__ c, int n) {
    const size_t base = (size_t)n * B_DIM * K_DIM;
    const int k    = (blockIdx.x * blockDim.x + threadIdx.x) * 4;
    const int row0 = blockIdx.y * ROWS;
    float4 rv = *(const float4*)(r + n * K_DIM + k);
    for (int rr = 0; rr < ROWS; ++rr) {
        int row = row0 + rr;
        float cb = c[n * B_DIM + row] * (float)B_DIM;
        size_t idx = base + (size_t)row * K_DIM + k;
        float4 e = *(const float4*)(E + idx);
        e.x *= rv.x * cb; e.y *= rv.y * cb; e.z *= rv.z * cb; e.w *= rv.w * cb;
        *(float4*)(E + idx) = e;
    }
}

extern "C" void kernel_launch(void* const* d_in, const int* in_sizes, int n_in,
                              void* d_out, int out_size, void* d_ws, size_t ws_size,
                              hipStream_t stream) {
    (void)in_sizes; (void)n_in; (void)out_size; (void)ws_size;
    const float* logits = (const float*)d_in[0];
    float* E  = (float*)d_out;                 // E materialized in output buffer
    float* ws = (float*)d_ws;
    unsigned* umax = (unsigned*)(ws + WS_UMAX);
    float* colsum  = ws + WS_COLSUM;
    float* r       = ws + WS_R;
    float* c       = ws + WS_C;
    float* rowdot  = ws + WS_ROWDOT;

    k_init<<<(WS_TOTAL + 255) / 256, 256, 0, stream>>>(ws);

    const dim3 gsweep(K_DIM / 4 / 256, B_DIM / ROWS);     // (4, 128)
    const int  gwmma = (B_DIM / 16) * KSPLIT / 8;         // 256 blocks of 8 waves

    for (int n = 0; n < NSLICE; ++n) {        // per-slice: 128 MB stays in L2
        k_max<<<2048, 256, 0, stream>>>(logits, umax, n);
        k_expcol<<<gsweep, 256, 0, stream>>>(logits, E, colsum, umax, n);
        for (int it = 0; it < 3; ++it) {
            if (it > 0) k_coldot<<<gsweep, 256, 0, stream>>>(E, c, colsum, n);
            k_r<<<K_DIM / 256, 256, 0, stream>>>(colsum, r, n);
            k_rowdot<<<gwmma, 256, 0, stream>>>(E, r, rowdot, n);
            k_c<<<B_DIM / 256, 256, 0, stream>>>(rowdot, c, n);
        }
        k_final<<<gsweep, 256, 0, stream>>>(E, r, c, n);
    }
}